// CausalGNN_69904887709752
// MI455X (gfx1250) — compile-verified
//
#include <hip/hip_runtime.h>

#define NN 50000
#define EE 800000
#define CH 128

typedef __attribute__((ext_vector_type(2))) float v2f;
typedef __attribute__((ext_vector_type(8))) float v8f;

// ---- degree / normalization -------------------------------------------------
__global__ void k_deg_init(float* __restrict__ deg) {
  int i = blockIdx.x * 256 + threadIdx.x;
  if (i < NN) deg[i] = 1.0f;               // self-loop contributes 1
}

__global__ void k_deg_count(const int* __restrict__ dst, float* __restrict__ deg) {
  int e = blockIdx.x * 256 + threadIdx.x;
  if (e < EE) atomicAdd(&deg[dst[e]], 1.0f);
}

__global__ void k_dinv(float* __restrict__ deg) {
  int i = blockIdx.x * 256 + threadIdx.x;
  if (i < NN) deg[i] = rsqrtf(deg[i]);      // deg >= 1 always
}

// ---- dense: OUT[16 x 128] = X_tile @ W^T via fp32 WMMA 16x16x4 -------------
__global__ void __launch_bounds__(256) k_gemm_xwt(const float* __restrict__ X,
                                                  const float* __restrict__ W,
                                                  float* __restrict__ OUT) {
  __shared__ float xs[CH * 16];             // transposed tile: xs[k*16 + row]
  const int tid  = threadIdx.x;
  const int row0 = blockIdx.x * 16;

  // cooperative transposed load of 16x128 tile (2048 floats, 8 per thread)
  #pragma unroll
  for (int it = 0; it < 8; ++it) {
    int t = tid + it * 256;
    int row = t >> 7, k = t & 127;
    xs[k * 16 + row] = X[(row0 + row) * CH + k];
  }
  __syncthreads();

  const int wave = tid >> 5;                // 0..7 -> channel tile
  const int lane = tid & 31;
  const int half = lane >> 4;               // K half-select per ISA layout
  const int lrow = lane & 15;               // A: M index / B: N index
  const float* wrow = W + (wave * 16 + lrow) * CH;  // W[o, :]

  v8f acc = {};
  #pragma unroll 4
  for (int k = 0; k < CH; k += 4) {
    const int kk = k + 2 * half;
    v2f a;                                  // A[M=lrow, K=kk..kk+1]
    a.x = xs[kk * 16 + lrow];
    a.y = xs[(kk + 1) * 16 + lrow];
    v2f b = *(const v2f*)(wrow + kk);       // B[K=kk..kk+1, N=lrow] = W[o, kk..kk+1]
    acc = __builtin_amdgcn_wmma_f32_16x16x4_f32(
        /*neg_a=*/false, a, /*neg_b=*/false, b,
        /*c_mod=*/(short)0, acc, /*reuse_a=*/false, /*reuse_b=*/false);
  }

  // D layout: VGPR r -> row (r + 8*half), col = wave*16 + lrow
  float* orow = OUT + (row0 + 8 * half) * CH + wave * 16 + lrow;
  #pragma unroll
  for (int r = 0; r < 8; ++r) orow[r * CH] = acc[r];
}

// ---- scatter-aggregate ------------------------------------------------------
__global__ void k_selfloop(const float* __restrict__ lin, const float* __restrict__ dinv,
                           float* __restrict__ agg) {
  int idx = blockIdx.x * 256 + threadIdx.x;      // N*CH threads
  int i = idx >> 7;
  float di = dinv[i];
  agg[idx] = lin[idx] * di * di;                 // self-loop term, also inits buffer
}

__global__ void k_scatter(const float* __restrict__ lin, const int* __restrict__ src,
                          const int* __restrict__ dst, const float* __restrict__ dinv,
                          float* __restrict__ agg) {
  int idx = blockIdx.x * 256 + threadIdx.x;      // E*CH threads
  int e = idx >> 7, c = idx & 127;
  int s = src[e], d = dst[e];
  float nrm = dinv[s] * dinv[d];
  atomicAdd(&agg[d * CH + c], lin[s * CH + c] * nrm);
}

__global__ void k_bias_relu(const float* __restrict__ agg, const float* __restrict__ b,
                            float* __restrict__ act) {
  int idx = blockIdx.x * 256 + threadIdx.x;      // N*CH threads
  int c = idx & 127;
  act[idx] = fmaxf(agg[idx] + b[c], 0.0f);
}

// ---- output head: sigmoid(h @ Wout^T + bout) -------------------------------
__global__ void k_head(const float* __restrict__ H, const float* __restrict__ Wout,
                       const float* __restrict__ bout, float* __restrict__ out) {
  int node = blockIdx.x * 8 + (threadIdx.x >> 5);
  int lane = threadIdx.x & 31;
  const float* h = H + node * CH;
  float s = 0.0f;
  #pragma unroll
  for (int i = 0; i < 4; ++i) s += h[lane + 32 * i] * Wout[lane + 32 * i];
  #pragma unroll
  for (int off = 16; off > 0; off >>= 1) s += __shfl_xor(s, off, 32);
  if (lane == 0) {
    float z = s + bout[0];
    out[node] = 1.0f / (1.0f + __expf(-z));
  }
}

extern "C" void kernel_launch(void* const* d_in, const int* in_sizes, int n_in,
                              void* d_out, int out_size, void* d_ws, size_t ws_size,
                              hipStream_t stream) {
  const float* x    = (const float*)d_in[0];
  const int*   ei   = (const int*)d_in[1];   // [2, E] flat: row0 = src, row1 = dst
  const float* W1   = (const float*)d_in[2];
  const float* b1   = (const float*)d_in[3];
  const float* W2   = (const float*)d_in[4];
  const float* b2   = (const float*)d_in[5];
  const float* Wout = (const float*)d_in[6];
  const float* bout = (const float*)d_in[7];
  const int* src = ei;
  const int* dst = ei + EE;
  float* out = (float*)d_out;

  // workspace carve (all fully rewritten every call -> deterministic replays)
  float* dinv = (float*)d_ws;                      // NN floats
  float* lin  = dinv + NN;                         // NN*CH floats (GEMM output)
  float* agg  = lin  + (size_t)NN * CH;            // NN*CH floats (accumulator)
  float* act  = agg  + (size_t)NN * CH;            // NN*CH floats (activations)

  dim3 B(256);
  // normalization (shared by both layers)
  k_deg_init <<<(NN + 255) / 256, B, 0, stream>>>(dinv);
  k_deg_count<<<(EE + 255) / 256, B, 0, stream>>>(dst, dinv);
  k_dinv     <<<(NN + 255) / 256, B, 0, stream>>>(dinv);

  // layer 1
  k_gemm_xwt <<<NN / 16,        B, 0, stream>>>(x, W1, lin);
  k_selfloop <<<NN * CH / 256,  B, 0, stream>>>(lin, dinv, agg);
  k_scatter  <<<EE * CH / 256,  B, 0, stream>>>(lin, src, dst, dinv, agg);
  k_bias_relu<<<NN * CH / 256,  B, 0, stream>>>(agg, b1, act);

  // layer 2
  k_gemm_xwt <<<NN / 16,        B, 0, stream>>>(act, W2, lin);
  k_selfloop <<<NN * CH / 256,  B, 0, stream>>>(lin, dinv, agg);
  k_scatter  <<<EE * CH / 256,  B, 0, stream>>>(lin, src, dst, dinv, agg);
  k_bias_relu<<<NN * CH / 256,  B, 0, stream>>>(agg, b2, act);

  // head
  k_head<<<NN / 8, B, 0, stream>>>(act, Wout, bout, out);
}